// conv_nonlinear_45715631898929
// MI455X (gfx1250) — compile-verified
//
#include <hip/hip_runtime.h>
#include <math.h>

typedef float v2f __attribute__((ext_vector_type(2)));
typedef float v8f __attribute__((ext_vector_type(8)));

#define WAVES     8
#define NTHREADS  (WAVES * 32)
#define NOUT      986           // (1000 - 15)/1 + 1
#define NK        64
#define HS        96            // activation row stride in floats (max padded width)
#define MROWS     32            // rows per wave (2 x 16-row WMMA sub-tiles sharing B)
#define NTILES    493           // 16*986 / 32, exact

// ---- LDS layout (float offsets) ----
// Weights stored TRANSPOSED: Wt[n][k], row stride WS = roundup4(din)+2 (even ->
// b64-aligned K pairs; gcd(WS,64)=2 -> bank-friendly column reads).
#define WS0 62
#define WS1 74
#define WS2 90
#define WS3 90
#define WS4 74
#define OFF_W0 0                        // 80*62 = 4960
#define OFF_W1 (OFF_W0 + 80*WS0)        // 96*74 = 7104
#define OFF_W2 (OFF_W1 + 96*WS1)        // 96*90 = 8640
#define OFF_W3 (OFF_W2 + 96*WS2)        // 80*90 = 7200
#define OFF_W4 (OFF_W3 + 80*WS3)        // 64*74 = 4736
#define OFF_B0 (OFF_W4 + 64*WS4)        // (80)
#define OFF_B1 (OFF_B0 + 80)            // (96)
#define OFF_B2 (OFF_B1 + 96)            // (96)
#define OFF_B3 (OFF_B2 + 96)            // (80)
#define OFF_B4 (OFF_B3 + 80)            // (64)
#define OFF_HW (OFF_B4 + 64)            // (64) head_w padded
#define OFF_HB (OFF_HW + 64)            // (1)  head bias
#define OFF_H  (((OFF_HB + 1) + 3) & ~3)        // 16B-aligned activation base
#define LDS_FLOATS (OFF_H + WAVES * MROWS * HS) // ~57700 floats = ~226 KB (<=320 KB/WG)

__device__ __forceinline__ void zero_lds(float* lds, int tot) {
    for (int idx = threadIdx.x; idx < tot; idx += NTHREADS)
        lds[idx] = 0.0f;
}

// Scatter-transpose W(k,[din,dout]) -> LDS Wt[dout(+pad)][WS]; global reads coalesced.
__device__ __forceinline__ void load_wT(const float* __restrict__ g, float* lds,
                                        int off, int din, int dout, int ws) {
    const int tot = din * dout;
    for (int idx = threadIdx.x; idx < tot; idx += NTHREADS) {
        int r = idx / dout;           // k (input) index
        int c = idx - r * dout;       // n (output) index
        lds[off + c * ws + r] = g[idx];
    }
}

__device__ __forceinline__ void load_bias(const float* __restrict__ g, float* lds,
                                          int off, int dout) {
    for (int idx = threadIdx.x; idx < dout; idx += NTHREADS)
        lds[off + idx] = g[idx];
}

// One layer, 32 rows: h(32 x KPAD, zero-padded) -> gelu(h @ W + b) (32 x NPAD), in place.
// Two 16-row A sub-tiles share each B fragment -> ~0.67 LDS b64 loads per WMMA.
// fp32 WMMA 16x16x4 fragments (ISA 7.12.2):
//   A: lane (m = lane&15), pair = h[m][k0+2*half], h[m][k0+2*half+1]
//   B: lane (n = t*16 + (lane&15)), pair = Wt[n][k0+2*half], Wt[n][k0+2*half+1]
//   C/D: element [v + 8*half][lane&15] in VGPR v.
template <int KPAD, int NPAD, int WSL>
__device__ __forceinline__ void layer_gemm(const float* __restrict__ lds,
                                           int wOff, int bOff, float* hbuf) {
    constexpr int NT = NPAD / 16;
    const int lane = threadIdx.x & 31;
    const int half = lane >> 4;
    const int nl   = lane & 15;

    const float* a0   = hbuf + nl * HS + 2 * half;         // rows 0..15
    const float* a1   = a0 + 16 * HS;                      // rows 16..31
    const float* wrow = lds + wOff + nl * WSL + 2 * half;  // B column, K-phase

    v8f acc0[NT], acc1[NT];
#pragma unroll
    for (int t = 0; t < NT; ++t) { acc0[t] = v8f{}; acc1[t] = v8f{}; }  // SRC2 = inline 0

    for (int k0 = 0; k0 < KPAD; k0 += 4) {
        v2f av0 = *(const v2f*)(a0 + k0);                  // ds_load_b64
        v2f av1 = *(const v2f*)(a1 + k0);                  // ds_load_b64
#pragma unroll
        for (int t = 0; t < NT; ++t) {
            v2f b = *(const v2f*)(wrow + t * 16 * WSL + k0);   // shared by both sub-tiles
            acc0[t] = __builtin_amdgcn_wmma_f32_16x16x4_f32(
                false, av0, false, b, (short)0, acc0[t], false, false);
            acc1[t] = __builtin_amdgcn_wmma_f32_16x16x4_f32(
                false, av1, false, b, (short)0, acc1[t], false, false);
        }
    }

    // epilogue: +bias, exact GELU, write back (all K reads done -> in-place OK)
#pragma unroll
    for (int t = 0; t < NT; ++t) {
        float bv = lds[bOff + t * 16 + nl];
#pragma unroll
        for (int v = 0; v < 8; ++v) {
            float x0 = acc0[t][v] + bv;
            float x1 = acc1[t][v] + bv;
            float g0 = 0.5f * x0 * (1.0f + erff(x0 * 0.70710678118654752f));
            float g1 = 0.5f * x1 * (1.0f + erff(x1 * 0.70710678118654752f));
            hbuf[(v + 8 * half) * HS + t * 16 + nl]        = g0;
            hbuf[(16 + v + 8 * half) * HS + t * 16 + nl]   = g1;
        }
    }
}

__global__ __launch_bounds__(NTHREADS, 1)
void conv_mlp_wmma_kernel(const float* __restrict__ x,
                          const float* __restrict__ w0, const float* __restrict__ b0,
                          const float* __restrict__ w1, const float* __restrict__ b1,
                          const float* __restrict__ w2, const float* __restrict__ b2,
                          const float* __restrict__ w3, const float* __restrict__ b3,
                          const float* __restrict__ w4, const float* __restrict__ b4,
                          const float* __restrict__ hw, const float* __restrict__ hb,
                          float* __restrict__ out) {
    extern __shared__ float lds[];
    const int k = blockIdx.y;

    // Zero entire weight/bias/head region (all pads become exact 0), then stage.
    zero_lds(lds, OFF_H);
    __syncthreads();
    load_wT(w0 + k * 60 * 72, lds, OFF_W0, 60, 72, WS0);
    load_wT(w1 + k * 72 * 86, lds, OFF_W1, 72, 86, WS1);
    load_wT(w2 + k * 86 * 86, lds, OFF_W2, 86, 86, WS2);
    load_wT(w3 + k * 86 * 71, lds, OFF_W3, 86, 71, WS3);
    load_wT(w4 + k * 71 * 59, lds, OFF_W4, 71, 59, WS4);
    load_bias(b0 + k * 72, lds, OFF_B0, 72);
    load_bias(b1 + k * 86, lds, OFF_B1, 86);
    load_bias(b2 + k * 86, lds, OFF_B2, 86);
    load_bias(b3 + k * 71, lds, OFF_B3, 71);
    load_bias(b4 + k * 59, lds, OFF_B4, 59);
    load_bias(hw, lds, OFF_HW, 59);
    if (threadIdx.x == 0) lds[OFF_HB] = hb[0];
    __syncthreads();

    const int wave  = threadIdx.x >> 5;
    const int wtile = blockIdx.x * WAVES + wave;   // 32-row tile index, 0..492
    if (wtile >= NTILES) return;                   // whole-wave exit: EXEC stays all-ones

    const int lane = threadIdx.x & 31;
    float* hbuf = lds + OFF_H + wave * (MROWS * HS);
    const int r0 = wtile * MROWS;                  // global row = b*986 + lo

    // Build layer-0 input rows: xw[r][c*15+j] = x[b, c, lo+j]   (32*60 = 1920 = 60*32)
    for (int idx = lane; idx < MROWS * 60; idx += 32) {
        int m  = idx / 60;
        int kk = idx - m * 60;
        int r  = r0 + m;
        int bb = r / NOUT;
        int lo = r - bb * NOUT;
        int c  = kk / 15;
        int j  = kk - c * 15;
        hbuf[m * HS + kk] = x[(bb * 4 + c) * 1000 + lo + j];
    }

    layer_gemm<60, 80, WS0>(lds, OFF_W0, OFF_B0, hbuf);   // 60 -> 72 (pad 80)
    layer_gemm<72, 96, WS1>(lds, OFF_W1, OFF_B1, hbuf);   // 72 -> 86 (pad 96)
    layer_gemm<88, 96, WS2>(lds, OFF_W2, OFF_B2, hbuf);   // 86 -> 86 (pad 96)
    layer_gemm<88, 80, WS3>(lds, OFF_W3, OFF_B3, hbuf);   // 86 -> 71 (pad 80)
    layer_gemm<72, 64, WS4>(lds, OFF_W4, OFF_B4, hbuf);   // 71 -> 59 (pad 64)

    // Head: out[b,k,lo] = h(59) . head_w + head_b   (all 32 lanes, one row each)
    {
        float s = lds[OFF_HB];
        for (int j = 0; j < 59; ++j)
            s += hbuf[lane * HS + j] * lds[OFF_HW + j];
        int r  = r0 + lane;
        int bb = r / NOUT;
        int lo = r - bb * NOUT;
        out[(bb * NK + k) * NOUT + lo] = s;
    }
}

extern "C" void kernel_launch(void* const* d_in, const int* in_sizes, int n_in,
                              void* d_out, int out_size, void* d_ws, size_t ws_size,
                              hipStream_t stream) {
    (void)in_sizes; (void)n_in; (void)d_ws; (void)ws_size; (void)out_size;
    const float* x  = (const float*)d_in[0];
    const float* w0 = (const float*)d_in[1];
    const float* b0 = (const float*)d_in[2];
    const float* w1 = (const float*)d_in[3];
    const float* b1 = (const float*)d_in[4];
    const float* w2 = (const float*)d_in[5];
    const float* b2 = (const float*)d_in[6];
    const float* w3 = (const float*)d_in[7];
    const float* b3 = (const float*)d_in[8];
    const float* w4 = (const float*)d_in[9];
    const float* b4 = (const float*)d_in[10];
    const float* hw = (const float*)d_in[11];
    const float* hb = (const float*)d_in[12];
    float* out = (float*)d_out;

    dim3 grid((NTILES + WAVES - 1) / WAVES, NK);   // (62, 64)
    size_t shmem = (size_t)LDS_FLOATS * sizeof(float);  // ~226 KB (<= 320 KB/WG on CDNA5)
    conv_mlp_wmma_kernel<<<grid, NTHREADS, shmem, stream>>>(
        x, w0, b0, w1, b1, w2, b2, w3, b3, w4, b4, hw, hb, out);
}